// AttentionLayer_21612275433590
// MI455X (gfx1250) — compile-verified
//
#include <hip/hip_runtime.h>
#include <math.h>

// Problem constants (from the reference)
#define B_  1024
#define P_  32
#define L_  64
#define D_  128
#define NW  4               // waves per block (128 threads)

// LDS layout (in floats)
#define TSTRIDE   132       // padded tile row stride: 528B, 16B aligned, bank-conflict-free
#define ATT_OFF   0         // attention[256] (a_u | a_z)
#define ACC_OFF   256       // weighted accumulator [128]
#define RED_OFF   384       // per-wave e_u partials [4]
#define ALPHA_OFF 392       // 4 waves * 64 alpha values
#define TILE_OFF  648       // 4 waves * 2 buffers * 64*132 tile floats
#define TILE_SZ   (L_ * TSTRIDE)                     // 8448 floats per buffer
#define LDS_FLOATS (TILE_OFF + NW * 2 * TILE_SZ)     // 68232 floats = 272,928 B (< 320 KB/WGP)

typedef __attribute__((ext_vector_type(2))) float v2f;
typedef __attribute__((ext_vector_type(8))) float v8f;
typedef __attribute__((ext_vector_type(4))) int   v4i;

#define AS_GLOBAL __attribute__((address_space(1)))
#define AS_LDS    __attribute__((address_space(3)))

__device__ __forceinline__ float wave_reduce_sum(float v) {
#pragma unroll
  for (int o = 16; o >= 1; o >>= 1) v += __shfl_xor(v, o, 32);
  return v;
}
__device__ __forceinline__ float wave_reduce_max(float v) {
#pragma unroll
  for (int o = 16; o >= 1; o >>= 1) v = fmaxf(v, __shfl_xor(v, o, 32));
  return v;
}

// Wait until all outstanding async (ASYNCcnt) LDS fills are complete.
__device__ __forceinline__ void async_wait0() {
#if __has_builtin(__builtin_amdgcn_s_wait_asynccnt)
  __builtin_amdgcn_s_wait_asynccnt(0);
  asm volatile("" ::: "memory");           // compiler barrier: no LDS reads hoisted above
#else
  asm volatile("s_wait_asynccnt 0x0" ::: "memory");
#endif
}

// One per-lane 16B async copy HBM -> LDS (GLOBAL_LOAD_ASYNC_TO_LDS_B128, ASYNCcnt).
__device__ __forceinline__ void async_copy16(const float* gp, float* lp) {
#if __has_builtin(__builtin_amdgcn_global_load_async_to_lds_b128)
  __builtin_amdgcn_global_load_async_to_lds_b128(
      (AS_GLOBAL v4i*)(uintptr_t)gp,           // AS1 v4i* source (per diagnostic)
      (AS_LDS v4i*)(unsigned)(uintptr_t)lp,    // AS3 pointers are 32-bit LDS offsets
      0, 0);
#else
  const unsigned d = (unsigned)(uintptr_t)lp;  // low 32 bits of flat addr = LDS byte offset
  asm volatile("global_load_async_to_lds_b128 %0, %1, off"
               :: "v"(d), "v"(gp) : "memory");
#endif
}

// Issue the full 64x128 fp32 tile (32 KB) as 64 async row-fills (512 B each).
__device__ __forceinline__ void issue_tile_async(const float* gsrc, float* tile, int lane) {
#pragma unroll 8
  for (int l = 0; l < L_; ++l)
    async_copy16(gsrc + l * D_ + lane * 4, tile + l * TSTRIDE + lane * 4);
}

__global__ __launch_bounds__(128)
void attn_fused(const float* __restrict__ hu, const float* __restrict__ hz,
                const long long* __restrict__ plen, const float* __restrict__ att,
                float* __restrict__ out) {
  __shared__ float lds[LDS_FLOATS];

  const int tid  = threadIdx.x;      // 0..127
  const int wave = tid >> 5;
  const int lane = tid & 31;
  const int b    = blockIdx.x;

  float* tbuf0 = &lds[TILE_OFF + wave * 2 * TILE_SZ];
  float* tbuf1 = tbuf0 + TILE_SZ;

  // Kick off the first tile's async fill immediately (wave-private LDS region).
  const float* gbase = hz + (((size_t)b * P_ + wave) * L_) * D_;
  issue_tile_async(gbase, tbuf0, lane);

  // Stage attention into LDS; zero the cross-wave accumulator.
  lds[ATT_OFF + tid]       = att[tid];        // a_u
  lds[ATT_OFF + 128 + tid] = att[128 + tid];  // a_z
  lds[ACC_OFF + tid] = 0.0f;

  // e_u = dot(hu[b,:], a_u): one MAC per thread, wave reduce, combine via LDS.
  float part = hu[b * D_ + tid] * att[tid];
  part = wave_reduce_sum(part);
  if (lane == 0) lds[RED_OFF + wave] = part;
  __syncthreads();
  const float e_u = lds[RED_OFF + 0] + lds[RED_OFF + 1] +
                    lds[RED_OFF + 2] + lds[RED_OFF + 3];

  float*       abuf = &lds[ALPHA_OFF + wave * 64];
  const float* az   = &lds[ATT_OFF + 128];

  // Persistent WMMA accumulators: 8 N-tiles of 16 -> D=128. Accumulate over k AND p.
  v8f cacc[8];
#pragma unroll
  for (int nt = 0; nt < 8; ++nt) cacc[nt] = {};

  const int n  = lane & 15;          // N index inside a 16-wide tile
  const int h2 = (lane & 16) >> 3;   // 0 for lanes 0-15, 2 for lanes 16-31 (K pairing)

  for (int pi = 0; pi < P_ / NW; ++pi) {
    const int p = wave + NW * pi;
    float* tile = (pi & 1) ? tbuf1 : tbuf0;

    // Complete the async fill of the buffer we are about to compute on,
    // then immediately start fetching the NEXT tile into the other buffer.
    async_wait0();
    if (pi + 1 < P_ / NW) {
      const float* gnext = hz + (((size_t)b * P_ + (p + NW)) * L_) * D_;
      issue_tile_async(gnext, (pi & 1) ? tbuf0 : tbuf1, lane);
    }

    // ---- e_z for rows `lane` and `lane+32` (dot with a_z, broadcast LDS reads) ----
    float e0 = 0.f, e1 = 0.f;
    const float* r0p = tile + lane * TSTRIDE;
    const float* r1p = tile + (lane + 32) * TSTRIDE;
#pragma unroll 8
    for (int d = 0; d < D_; d += 4) {
      const float4 a4 = *(const float4*)(az + d);
      const float4 x0 = *(const float4*)(r0p + d);
      const float4 x1 = *(const float4*)(r1p + d);
      e0 += x0.x * a4.x + x0.y * a4.y + x0.z * a4.z + x0.w * a4.w;
      e1 += x1.x * a4.x + x1.y * a4.y + x1.z * a4.z + x1.w * a4.w;
    }

    // leaky_relu(e_u + e_z, 0.2), then mask l >= path_len with -inf
    e0 += e_u; e1 += e_u;
    e0 = e0 > 0.f ? e0 : 0.2f * e0;
    e1 = e1 > 0.f ? e1 : 0.2f * e1;
    const int pl = (int)plen[b * P_ + p];   // int64 input
    if (lane      >= pl) e0 = -INFINITY;
    if (lane + 32 >= pl) e1 = -INFINITY;

    // softmax over L=64 (2 values per lane)
    const float m   = wave_reduce_max(fmaxf(e0, e1));
    const float x0e = expf(e0 - m);
    const float x1e = expf(e1 - m);
    const float s   = wave_reduce_sum(x0e + x1e);
    const float inv = 1.0f / s;
    abuf[lane]      = x0e * inv;
    abuf[lane + 32] = x1e * inv;   // wave-private LDS; DS ops are in-order per wave

    // ---- weighted(1x128) += alpha(1x64) * tile(64x128) via V_WMMA_F32_16X16X4_F32 ----
    // A (16x4 f32): replicate alpha chunk over all M rows:
    //   VGPR0 = alpha[4k + (lane<16 ? 0 : 2)], VGPR1 = alpha[4k + (lane<16 ? 1 : 3)]
    // B (4x16 f32): VGPR0 = tile[4k+{0|2}, nt*16+n], VGPR1 = tile[4k+{1|3}, nt*16+n]
#pragma unroll 2
    for (int k = 0; k < 16; ++k) {
      const int r0 = 4 * k + h2;
      v2f A;
      A.x = abuf[r0];
      A.y = abuf[r0 + 1];
#pragma unroll
      for (int nt = 0; nt < 8; ++nt) {
        v2f Bv;
        Bv.x = tile[r0 * TSTRIDE + nt * 16 + n];
        Bv.y = tile[(r0 + 1) * TSTRIDE + nt * 16 + n];
        cacc[nt] = __builtin_amdgcn_wmma_f32_16x16x4_f32(
            false, A, false, Bv, (short)0, cacc[nt], false, false);
      }
    }
  }

  // Row M=0 of C/D lives in VGPR0, lanes 0..15 (N = lane). Combine waves via LDS atomics.
  if (lane < 16) {
#pragma unroll
    for (int nt = 0; nt < 8; ++nt)
      atomicAdd(&lds[ACC_OFF + nt * 16 + lane], cacc[nt][0]);
  }
  __syncthreads();

  // out = hu + elu(weighted)
  const float w  = lds[ACC_OFF + tid];
  const float el = w > 0.f ? w : (expf(w) - 1.0f);
  out[(size_t)b * D_ + tid] = hu[b * D_ + tid] + el;
}

extern "C" void kernel_launch(void* const* d_in, const int* in_sizes, int n_in,
                              void* d_out, int out_size, void* d_ws, size_t ws_size,
                              hipStream_t stream) {
  (void)in_sizes; (void)n_in; (void)out_size; (void)d_ws; (void)ws_size;
  const float*     hu   = (const float*)d_in[0];
  const float*     hz   = (const float*)d_in[1];
  const long long* plen = (const long long*)d_in[2];  // int64 path_lengths
  const float*     att  = (const float*)d_in[3];
  float*           out  = (float*)d_out;
  attn_fused<<<dim3(B_), dim3(128), 0, stream>>>(hu, hz, plen, att, out);
}